// LengthRegulator_23948737642960
// MI455X (gfx1250) — compile-verified
//
#include <hip/hip_runtime.h>
#include <hip/hip_bf16.h>

#define T_PHONE 1024
#define TILE_T  256
#define CPB     8   // channels per block in the gather kernel

typedef __attribute__((ext_vector_type(4))) int v4i;

// ---------------------------------------------------------------------------
// CDNA5 async global->LDS path (ASYNCcnt). Guarded so a missing builtin falls
// back to plain vector loads instead of breaking the compile.
// ---------------------------------------------------------------------------
#if defined(__HIP_DEVICE_COMPILE__) && defined(__gfx1250__)
#if __has_builtin(__builtin_amdgcn_global_load_async_to_lds_b128) && \
    __has_builtin(__builtin_amdgcn_s_wait_asynccnt)
#define LR_USE_ASYNC 1
#endif
#endif

// ---------------------------------------------------------------------------
// Kernel 1: per-batch inclusive cumsum of durations (T_PHONE = 1024).
// One 256-thread block per batch, 4 elements per thread, LDS scan.
// ---------------------------------------------------------------------------
__global__ void lr_cumsum(const int* __restrict__ dur, int* __restrict__ cum) {
    const int b   = blockIdx.x;
    const int tid = threadIdx.x;

    const int4* src = (const int4*)(dur + (size_t)b * T_PHONE);
    int4 v = src[tid];
    const int s0 = v.x;
    const int s1 = s0 + v.y;
    const int s2 = s1 + v.z;
    const int s3 = s2 + v.w;

    __shared__ int sums[256];
    sums[tid] = s3;

    // Hillis-Steele inclusive scan over the 256 per-thread sums.
    for (int off = 1; off < 256; off <<= 1) {
        __syncthreads();
        const int t = (tid >= off) ? sums[tid - off] : 0;
        __syncthreads();
        sums[tid] += t;
    }
    __syncthreads();

    const int prefix = (tid > 0) ? sums[tid - 1] : 0;
    int4 o;
    o.x = prefix + s0;
    o.y = prefix + s1;
    o.z = prefix + s2;
    o.w = prefix + s3;
    ((int4*)(cum + (size_t)b * T_PHONE))[tid] = o;
}

// ---------------------------------------------------------------------------
// Kernel 2: per-frame phone index via upper-bound binary search in LDS.
// Stages cum[b][0..1023] (4KB) into LDS with async-to-LDS b128 copies
// (ASYNCcnt path), then each thread runs a branchless 10-step search.
// p[b,t] = clip(searchsorted(cum, t, 'right'), 0, T-1) if valid else -1.
// Also writes mel_mask (= !valid) as float into the output tail.
// ---------------------------------------------------------------------------
__global__ void lr_index(const int* __restrict__ cum, int* __restrict__ p,
                         float* __restrict__ mask, int maxF) {
    const int b   = blockIdx.y;
    const int tid = threadIdx.x;

    __shared__ int lcum[T_PHONE];
    const int* gsrc = cum + (size_t)b * T_PHONE;

#if LR_USE_ASYNC
    // 256 lanes x 16B = 4KB: one async b128 per lane, straight into LDS.
    __builtin_amdgcn_global_load_async_to_lds_b128(
        (__attribute__((address_space(1))) v4i*)(gsrc + tid * 4),
        (__attribute__((address_space(3))) v4i*)(lcum + tid * 4),
        0, 0);
    __builtin_amdgcn_s_wait_asynccnt(0);
#else
    ((int4*)lcum)[tid] = ((const int4*)gsrc)[tid];
#endif
    __syncthreads();

    const int t = blockIdx.x * TILE_T + tid;
    if (t < maxF) {
        const int total = lcum[T_PHONE - 1];
        // Branchless upper-bound over a power-of-two-sized sorted array;
        // result = min(#elements <= t, T_PHONE-1), which matches the
        // reference's searchsorted(..., 'right') followed by clip().
        int pos = 0;
#pragma unroll
        for (int step = T_PHONE >> 1; step >= 1; step >>= 1) {
            if (lcum[pos + step - 1] <= t) pos += step;
        }
        const bool valid = (t < total);
        p[(size_t)b * maxF + t] = valid ? pos : -1;
        // mel_mask is a ~452KB write-once stream: non-temporal store.
        __builtin_nontemporal_store(valid ? 0.0f : 1.0f,
                                    mask + (size_t)b * maxF + t);
    }
}

// ---------------------------------------------------------------------------
// Kernel 3: gather. Each thread owns one frame t, keeps p[b,t] in a register
// and streams CPB channels (coalesced load of x, coalesced store of out).
// Stores are non-temporal: out (~226 MiB) exceeds the 192MB L2, and keeping
// it out of L2 preserves residency for x (64 MiB, re-read ~3.5x) and p.
// ---------------------------------------------------------------------------
__global__ void lr_gather(const float* __restrict__ x, const int* __restrict__ p,
                          float* __restrict__ out, int maxF, int C) {
    const int t = blockIdx.x * TILE_T + threadIdx.x;
    if (t >= maxF) return;
    const int b  = blockIdx.z;
    const int c0 = blockIdx.y * CPB;

    const int idx = p[(size_t)b * maxF + t];

    const float* xb = x   + (size_t)b * C * T_PHONE;
    float*       ob = out + (size_t)b * C * (size_t)maxF;

#pragma unroll
    for (int i = 0; i < CPB; ++i) {
        const int c = c0 + i;
        const float v = (idx >= 0) ? xb[(size_t)c * T_PHONE + idx] : 0.0f;
        __builtin_nontemporal_store(v, ob + (size_t)c * maxF + t);
    }
}

// ---------------------------------------------------------------------------
// Host launcher
// ---------------------------------------------------------------------------
extern "C" void kernel_launch(void* const* d_in, const int* in_sizes, int n_in,
                              void* d_out, int out_size, void* d_ws, size_t ws_size,
                              hipStream_t stream) {
    const float* x   = (const float*)d_in[0];
    const int*   dur = (const int*)d_in[1];

    const int T = T_PHONE;
    const int n_x = in_sizes[0];        // B*C*T
    const int n_d = in_sizes[1];        // B*T
    const int C   = n_x / n_d;          // 512
    const int B   = n_d / T;            // 32
    // out_size = B*maxF*C (out) + B*maxF (mel_mask)
    const int maxF = out_size / (B * (C + 1));

    float* out  = (float*)d_out;
    float* mask = out + (size_t)B * C * (size_t)maxF;

    int* cum = (int*)d_ws;              // B*T ints
    int* p   = cum + (size_t)B * T;     // B*maxF ints

    lr_cumsum<<<B, 256, 0, stream>>>(dur, cum);

    dim3 g2((maxF + TILE_T - 1) / TILE_T, B);
    lr_index<<<g2, TILE_T, 0, stream>>>(cum, p, mask, maxF);

    dim3 g3((maxF + TILE_T - 1) / TILE_T, (C + CPB - 1) / CPB, B);
    lr_gather<<<g3, TILE_T, 0, stream>>>(x, p, out, maxF, C);
}